// SS3DBlock_59794534695125
// MI455X (gfx1250) — compile-verified
//
#include <hip/hip_runtime.h>
#include <math.h>

// ---- problem constants ----
#define B_SZ   2
#define T_B    16
#define H_S    24
#define W_S    24
#define DIM    96
#define DI     192      // D_INNER
#define NST    16       // D_STATE
#define RNK    6        // DT_RANK
#define KG     4        // K_GRP
#define LTOT   (T_B*H_S*W_S)   // 9216
#define CCH    (RNK + 2*NST)   // 38
#define NCH    96       // chunks
#define CHLEN  96       // steps per chunk  (96*96 = 9216)
#define LN_EPS 1e-5f
#define PFD    8        // scan prefetch distance (steps)

typedef __attribute__((ext_vector_type(2))) float v2f;
typedef __attribute__((ext_vector_type(8))) float v8f;

__device__ __forceinline__ float silu_f(float v) { return v / (1.0f + __expf(-v)); }
__device__ __forceinline__ float softplus_f(float v) {
    return (v > 20.0f) ? v : log1pf(expf(v));
}
// spectral order l = (h*W + w)*T + t  ->  spatial order l = t*H*W + h*W + w
__device__ __forceinline__ int spe2spa(int l) {
    int t = l & (T_B - 1);
    int hw = l >> 4;             // T_B == 16
    return t * (H_S * W_S) + hw;
}

// ---------------- GEMM 1: xz = x @ W_in, fused conv-silu / silu-gate ----------------
// One wave computes a 16x64 tile (4 accumulators): A fragment reused 4x.
__global__ __launch_bounds__(32)
void k_inproj(const float* __restrict__ x, const float* __restrict__ Win,
              const float* __restrict__ convw, const float* __restrict__ convb,
              float* __restrict__ xc, float* __restrict__ zb)
{
    const int row0 = blockIdx.x * 16;
    const int col0 = blockIdx.y * 64;
    const int lane = threadIdx.x;
    const int half = lane >> 4;
    const int lr   = lane & 15;

    const float* arow = x + (size_t)(row0 + lr) * DIM;
    v8f acc[4];
#pragma unroll
    for (int j = 0; j < 4; ++j) acc[j] = (v8f){};

#pragma unroll 2
    for (int k0 = 0; k0 < DIM; k0 += 4) {
        v2f a;
        a.x = arow[k0 + 2*half + 0];
        a.y = arow[k0 + 2*half + 1];
        const float* b0 = Win + (size_t)(k0 + 2*half + 0) * (2*DI) + col0 + lr;
        const float* b1 = Win + (size_t)(k0 + 2*half + 1) * (2*DI) + col0 + lr;
#pragma unroll
        for (int j = 0; j < 4; ++j) {
            v2f b;
            b.x = b0[j * 16];
            b.y = b1[j * 16];
            acc[j] = __builtin_amdgcn_wmma_f32_16x16x4_f32(false, a, false, b, (short)0, acc[j], false, false);
        }
    }
#pragma unroll
    for (int j = 0; j < 4; ++j) {
#pragma unroll
        for (int i = 0; i < 8; ++i) {
            int gr  = row0 + i + 8*half;
            int col = col0 + j * 16 + lr;
            float v = acc[j][i];
            if (col < DI) {
                float t = v * convw[col] + convb[col];
                xc[(size_t)gr * DI + col] = silu_f(t);
            } else {
                zb[(size_t)gr * DI + (col - DI)] = silu_f(v);
            }
        }
    }
}

// ---------------- GEMM 2: G[b][k] = X_layout(k) @ W_k^T   (L x 192) @ (192 x 38) ----------------
// One wave computes a 16x48 tile (3 accumulators, cols >= 38 masked).
__global__ __launch_bounds__(32)
void k_xproj(const float* __restrict__ xc, const float* __restrict__ xpw,
             float* __restrict__ G)
{
    const int bk = blockIdx.y;
    const int b  = bk >> 2, k = bk & 3;
    const int row0 = blockIdx.x * 16;
    const int lane = threadIdx.x;
    const int half = lane >> 4;
    const int lr   = lane & 15;

    int l   = row0 + lr;
    int pos = (k < 2) ? spe2spa(l) : l;
    const float* arow = xc + ((size_t)b * LTOT + pos) * DI;
    const float* Wk   = xpw + (size_t)k * CCH * DI;

    v8f acc[3];
#pragma unroll
    for (int j = 0; j < 3; ++j) acc[j] = (v8f){};

#pragma unroll 2
    for (int k0 = 0; k0 < DI; k0 += 4) {
        v2f a;
        a.x = arow[k0 + 2*half + 0];
        a.y = arow[k0 + 2*half + 1];
#pragma unroll
        for (int j = 0; j < 3; ++j) {
            int col = j * 16 + lr;
            v2f bv;
            if (col < CCH) {
                bv.x = Wk[(size_t)col * DI + k0 + 2*half + 0];
                bv.y = Wk[(size_t)col * DI + k0 + 2*half + 1];
            } else { bv.x = 0.0f; bv.y = 0.0f; }
            acc[j] = __builtin_amdgcn_wmma_f32_16x16x4_f32(false, a, false, bv, (short)0, acc[j], false, false);
        }
    }
#pragma unroll
    for (int j = 0; j < 3; ++j) {
        int col = j * 16 + lr;
        if (col < CCH) {
#pragma unroll
            for (int i = 0; i < 8; ++i) {
                int rl = row0 + i + 8*half;
                G[((size_t)(b*KG + k) * LTOT + rl) * CCH + col] = acc[j][i];
            }
        }
    }
}

// ---------------- Scan phase 1: per-chunk local scan -> end state + sum(dt) ----------------
__global__ __launch_bounds__(DI)
void k_scan_chunk(const float* __restrict__ G, const float* __restrict__ xc,
                  const float* __restrict__ dtw, const float* __restrict__ dtb,
                  const float* __restrict__ Alog,
                  float* __restrict__ S, float* __restrict__ dtsum)
{
    const int c = blockIdx.x, k = blockIdx.y, b = blockIdx.z;
    const int d = threadIdx.x;

    float Av[NST], wdt[RNK];
#pragma unroll
    for (int n = 0; n < NST; ++n) Av[n] = -__expf(Alog[(size_t)(k*DI + d) * NST + n]);
#pragma unroll
    for (int r = 0; r < RNK; ++r) wdt[r] = dtw[(size_t)(k*DI + d) * RNK + r];
    const float bias = dtb[k*DI + d];

    float h[NST];
#pragma unroll
    for (int n = 0; n < NST; ++n) h[n] = 0.0f;
    float dts = 0.0f;

    const float* Gb = G + (size_t)(b*KG + k) * LTOT * CCH;
    const float* xb = xc + (size_t)b * LTOT * DI;
    const int dir = (k & 1) ? -1 : 1;
    const int s0  = c * CHLEN;

    for (int t = 0; t < CHLEN; ++t) {
        int s = s0 + t;
        int l = (dir < 0) ? (LTOT - 1 - s) : s;
        // prefetch PFD steps ahead along the scan stream (speculative)
        {
            int sp = s + PFD;
            if (sp < LTOT) {
                int lp = (dir < 0) ? (LTOT - 1 - sp) : sp;
                __builtin_prefetch(Gb + (size_t)lp * CCH, 0, 1);
                int pp = (k < 2) ? spe2spa(lp) : lp;
                __builtin_prefetch(xb + (size_t)pp * DI + d, 0, 1);
            }
        }
        const float* g = Gb + (size_t)l * CCH;
        float dtr = bias;
#pragma unroll
        for (int r = 0; r < RNK; ++r) dtr += g[r] * wdt[r];
        float dt = softplus_f(dtr);
        dts += dt;
        int pos = (k < 2) ? spe2spa(l) : l;
        float du = dt * xb[(size_t)pos * DI + d];
#pragma unroll
        for (int n = 0; n < NST; ++n)
            h[n] = __expf(dt * Av[n]) * h[n] + du * g[RNK + n];
    }
    size_t base = (((size_t)(b*KG + k) * NCH + c) * DI + d) * NST;
#pragma unroll
    for (int n = 0; n < NST; ++n) S[base + n] = h[n];
    dtsum[((size_t)(b*KG + k) * NCH + c) * DI + d] = dts;
}

// ---------------- Scan phase 2: sequential carry over chunks (per b,k,d,n) ----------------
__global__ __launch_bounds__(256)
void k_scan_carry(const float* __restrict__ Alog, const float* __restrict__ dtsum,
                  const float* __restrict__ S, float* __restrict__ Hin)
{
    int idx = blockIdx.x * 256 + threadIdx.x;          // [0, B*K*DI*NST)
    int n  = idx & (NST - 1);
    int d  = (idx >> 4) % DI;
    int bk = idx / (DI * NST);
    int k  = bk & 3;
    float An = -__expf(Alog[(size_t)(k*DI + d) * NST + n]);
    float h = 0.0f;
    for (int c = 0; c < NCH; ++c) {
        size_t sb = ((size_t)bk * NCH + c) * DI;
        Hin[(sb + d) * NST + n] = h;
        float ds = dtsum[sb + d];
        h = __expf(An * ds) * h + S[(sb + d) * NST + n];
    }
}

// ---------------- Scan phase 3: re-apply with carry-in, emit y (atomic accumulate) ----------------
__global__ __launch_bounds__(DI)
void k_scan_apply(const float* __restrict__ G, const float* __restrict__ xc,
                  const float* __restrict__ dtw, const float* __restrict__ dtb,
                  const float* __restrict__ Alog, const float* __restrict__ Dsv,
                  const float* __restrict__ Hin, float* __restrict__ yacc)
{
    const int c = blockIdx.x, k = blockIdx.y, b = blockIdx.z;
    const int d = threadIdx.x;

    float Av[NST], wdt[RNK];
#pragma unroll
    for (int n = 0; n < NST; ++n) Av[n] = -__expf(Alog[(size_t)(k*DI + d) * NST + n]);
#pragma unroll
    for (int r = 0; r < RNK; ++r) wdt[r] = dtw[(size_t)(k*DI + d) * RNK + r];
    const float bias = dtb[k*DI + d];
    const float Dp   = Dsv[k*DI + d];

    float h[NST];
    size_t hb = (((size_t)(b*KG + k) * NCH + c) * DI + d) * NST;
#pragma unroll
    for (int n = 0; n < NST; ++n) h[n] = Hin[hb + n];

    const float* Gb = G + (size_t)(b*KG + k) * LTOT * CCH;
    const float* xb = xc + (size_t)b * LTOT * DI;
    float* yb = yacc + (size_t)b * LTOT * DI;
    const int dir = (k & 1) ? -1 : 1;
    const int s0  = c * CHLEN;

    for (int t = 0; t < CHLEN; ++t) {
        int s = s0 + t;
        int l = (dir < 0) ? (LTOT - 1 - s) : s;
        {
            int sp = s + PFD;
            if (sp < LTOT) {
                int lp = (dir < 0) ? (LTOT - 1 - sp) : sp;
                __builtin_prefetch(Gb + (size_t)lp * CCH, 0, 1);
                int pp = (k < 2) ? spe2spa(lp) : lp;
                __builtin_prefetch(xb + (size_t)pp * DI + d, 0, 1);
            }
        }
        const float* g = Gb + (size_t)l * CCH;
        float dtr = bias;
#pragma unroll
        for (int r = 0; r < RNK; ++r) dtr += g[r] * wdt[r];
        float dt = softplus_f(dtr);
        int pos = (k < 2) ? spe2spa(l) : l;
        float u  = xb[(size_t)pos * DI + d];
        float du = dt * u;
        float y = 0.0f;
#pragma unroll
        for (int n = 0; n < NST; ++n) {
            h[n] = __expf(dt * Av[n]) * h[n] + du * g[RNK + n];
            y += h[n] * g[RNK + NST + n];
        }
        y += Dp * u;
        atomicAdd(&yb[(size_t)pos * DI + d], y);
    }
}

// ---------------- LayerNorm + z-gate (in place over zb) ----------------
__global__ __launch_bounds__(DI)
void k_norm(const float* __restrict__ yacc, float* __restrict__ zb,
            const float* __restrict__ gamma, const float* __restrict__ beta)
{
    const int row = blockIdx.x;
    const int d = threadIdx.x;
    float v = yacc[(size_t)row * DI + d];
    float s = v, s2 = v * v;
#pragma unroll
    for (int off = 16; off > 0; off >>= 1) {
        s  += __shfl_down(s, off, 32);
        s2 += __shfl_down(s2, off, 32);
    }
    __shared__ float r0[6], r1[6];
    __shared__ float mu_s, rinv_s;
    int lane = d & 31, wid = d >> 5;
    if (lane == 0) { r0[wid] = s; r1[wid] = s2; }
    __syncthreads();
    if (d == 0) {
        float ts = 0.0f, ts2 = 0.0f;
#pragma unroll
        for (int w = 0; w < 6; ++w) { ts += r0[w]; ts2 += r1[w]; }
        float mu = ts / (float)DI;
        float var = ts2 / (float)DI - mu * mu;
        mu_s = mu;
        rinv_s = rsqrtf(var + LN_EPS);
    }
    __syncthreads();
    float gn = ((v - mu_s) * rinv_s) * gamma[d] + beta[d];
    zb[(size_t)row * DI + d] = gn * zb[(size_t)row * DI + d];
}

// ---------------- GEMM 3: out = gated @ W_out   [B*L,192] @ [192,96] ----------------
// One wave computes a full 16x96 row-block (6 accumulators).
__global__ __launch_bounds__(32)
void k_outproj(const float* __restrict__ gbuf, const float* __restrict__ Wout,
               float* __restrict__ out)
{
    const int row0 = blockIdx.x * 16;
    const int lane = threadIdx.x;
    const int half = lane >> 4;
    const int lr   = lane & 15;
    const float* arow = gbuf + (size_t)(row0 + lr) * DI;

    v8f acc[6];
#pragma unroll
    for (int j = 0; j < 6; ++j) acc[j] = (v8f){};

#pragma unroll 2
    for (int k0 = 0; k0 < DI; k0 += 4) {
        v2f a;
        a.x = arow[k0 + 2*half + 0];
        a.y = arow[k0 + 2*half + 1];
        const float* b0 = Wout + (size_t)(k0 + 2*half + 0) * DIM + lr;
        const float* b1 = Wout + (size_t)(k0 + 2*half + 1) * DIM + lr;
#pragma unroll
        for (int j = 0; j < 6; ++j) {
            v2f b;
            b.x = b0[j * 16];
            b.y = b1[j * 16];
            acc[j] = __builtin_amdgcn_wmma_f32_16x16x4_f32(false, a, false, b, (short)0, acc[j], false, false);
        }
    }
#pragma unroll
    for (int j = 0; j < 6; ++j) {
#pragma unroll
        for (int i = 0; i < 8; ++i) {
            int gr = row0 + i + 8*half;
            out[(size_t)gr * DIM + j * 16 + lr] = acc[j][i];
        }
    }
}

extern "C" void kernel_launch(void* const* d_in, const int* in_sizes, int n_in,
                              void* d_out, int out_size, void* d_ws, size_t ws_size,
                              hipStream_t stream)
{
    const float* x     = (const float*)d_in[0];
    const float* Win   = (const float*)d_in[1];
    const float* convw = (const float*)d_in[2];
    const float* convb = (const float*)d_in[3];
    const float* xpw   = (const float*)d_in[4];
    const float* dtw   = (const float*)d_in[5];
    const float* dtb   = (const float*)d_in[6];
    const float* Alog  = (const float*)d_in[7];
    const float* Dsv   = (const float*)d_in[8];
    const float* gamma = (const float*)d_in[9];
    const float* beta  = (const float*)d_in[10];
    const float* Wout  = (const float*)d_in[11];
    float* out = (float*)d_out;

    float* ws = (float*)d_ws;
    size_t o = 0;
    float* xc    = ws + o; o += (size_t)B_SZ * LTOT * DI;
    float* zb    = ws + o; o += (size_t)B_SZ * LTOT * DI;
    float* G     = ws + o; o += (size_t)B_SZ * KG * LTOT * CCH;
    float* yacc  = ws + o; o += (size_t)B_SZ * LTOT * DI;
    float* S     = ws + o; o += (size_t)B_SZ * KG * NCH * DI * NST;
    float* dtsum = ws + o; o += (size_t)B_SZ * KG * NCH * DI;
    float* Hin   = ws + o; o += (size_t)B_SZ * KG * NCH * DI * NST;
    (void)o; (void)ws_size; (void)in_sizes; (void)n_in; (void)out_size;

    const int BL = B_SZ * LTOT;                       // 18432

    k_inproj<<<dim3(BL/16, (2*DI)/64), 32, 0, stream>>>(x, Win, convw, convb, xc, zb);
    k_xproj<<<dim3(LTOT/16, B_SZ*KG), 32, 0, stream>>>(xc, xpw, G);
    k_scan_chunk<<<dim3(NCH, KG, B_SZ), DI, 0, stream>>>(G, xc, dtw, dtb, Alog, S, dtsum);
    k_scan_carry<<<(B_SZ*KG*DI*NST)/256, 256, 0, stream>>>(Alog, dtsum, S, Hin);
    hipMemsetAsync(yacc, 0, sizeof(float) * (size_t)B_SZ * LTOT * DI, stream);
    k_scan_apply<<<dim3(NCH, KG, B_SZ), DI, 0, stream>>>(G, xc, dtw, dtb, Alog, Dsv, Hin, yacc);
    k_norm<<<BL, DI, 0, stream>>>(yacc, zb, gamma, beta);
    k_outproj<<<BL/16, 32, 0, stream>>>(zb, Wout, out);
}